// RNN_41223096107537
// MI455X (gfx1250) — compile-verified
//
#include <hip/hip_runtime.h>
#include <hip/hip_bf16.h>

// Problem constants from the reference
#define E_DIM 768
#define O_DIM 3
#define T_LEN 32768

typedef __attribute__((ext_vector_type(2))) float v2f;
typedef __attribute__((ext_vector_type(4))) float v4f;
typedef __attribute__((ext_vector_type(8))) float v8f;

// ---------------------------------------------------------------------------
// Kernel 1: a[t] = dot(document[t, 0:768], w1[0, 0:768])  for all t.
// Memory-bound: 100.7 MB streamed once ~4.3us at 23.3 TB/s; whole document
// fits in the 192MB L2. V_WMMA_F32_16X16X4_F32 keeps the fp32 dots exact on
// the matrix pipe; wasted columns are free at this arithmetic intensity.
//
// Layouts (CDNA5 ISA 7.12.2, wave32):
//   A (16x4 f32): lanes 0-15 -> rows, VGPR0/1 = K0/K1; lanes 16-31 = K2/K3.
//   B (4x16 f32): VGPR r, lanes 0-15 -> K=r; lanes 16-31 -> K=2+r.
//   -> per lane both A and B load columns  kb + (lane>>4)*2 .. +1.
//   C/D (16x16 f32): VGPR r, lanes 0-15 -> M=r; lanes 16-31 -> M=8+r.
// B broadcasts the weight chunk into all 16 columns, so every D column holds
// the 16 dot products; lane 0 / lane 16 store them.
// ---------------------------------------------------------------------------
__global__ __launch_bounds__(256) void rnn_gemv_wmma(
    const float* __restrict__ doc,   // (T, 768)
    const float* __restrict__ w1,    // (1, 771); first 768 are w1x
    float* __restrict__ a_out)       // (T)
{
  const int lane = threadIdx.x & 31;
  const int wave = threadIdx.x >> 5;
  const int rb   = blockIdx.x * 8 + wave;     // 16-row block index
  const int m0   = rb * 16;
  const int half = lane >> 4;                 // 0: K0/K1 half, 1: K2/K3 half
  const int r    = lane & 15;

  const float* arow = doc + (size_t)(m0 + r) * E_DIM + half * 2;
  const float* wcol = w1 + half * 2;

  v8f acc = {0.f, 0.f, 0.f, 0.f, 0.f, 0.f, 0.f, 0.f};

#pragma unroll 8
  for (int kb = 0; kb < E_DIM; kb += 4) {
    v2f av = *(const v2f*)(arow + kb);   // 8B-aligned (row stride 3072B)
    v2f bv = *(const v2f*)(wcol + kb);
    // 8 args: (neg_a, A, neg_b, B, c_mod, C, reuse_a, reuse_b)
    acc = __builtin_amdgcn_wmma_f32_16x16x4_f32(
        false, av, false, bv, (short)0, acc, false, false);
  }

  // All 16 columns of D identical: lane 0 -> rows 0..7, lane 16 -> rows 8..15.
  if ((lane & 15) == 0) {
    float* dst = a_out + m0 + half * 8;
#pragma unroll
    for (int i = 0; i < 8; ++i) dst[i] = acc[i];
  }
}

// ---------------------------------------------------------------------------
// Kernel 2: inherently serial nonlinear recurrence (sigmoid => no parallel
// scan). Critical path per step minimized to 4 dependent ops:
//     g -> v_exp_f32 -> v_add -> v_rcp_f32 -> v_fma -> g
// by expanding  v.prev_{t+1} = vc + (v.u) h_t + (vM).p_t  (the p_t term has a
// full step of slack -> off-path) and folding the exp2 base conversion into
// the same fma. a[t] loads feed only off-path terms and are double-buffered
// as aligned b128 loads two bodies ahead.
// ---------------------------------------------------------------------------
__global__ void rnn_serial(const float* __restrict__ a,    // (T) = w1x . x_t
                           const float* __restrict__ w1,   // (1, 771)
                           const float* __restrict__ b1p,  // (1)
                           const float* __restrict__ w2,   // (3, 4)
                           const float* __restrict__ b2,   // (3)
                           float* __restrict__ out)        // (3, 1)
{
  if (threadIdx.x != 0 || blockIdx.x != 0) return;

  const float v0 = w1[E_DIM + 0], v1 = w1[E_DIM + 1], v2 = w1[E_DIM + 2];
  const float bb1 = b1p[0];

  // w2 = [u | M], b2 = c
  const float u0 = w2[0], M00 = w2[1],  M01 = w2[2],  M02 = w2[3];
  const float u1 = w2[4], M10 = w2[5],  M11 = w2[6],  M12 = w2[7];
  const float u2 = w2[8], M20 = w2[9],  M21 = w2[10], M22 = w2[11];
  const float c0 = b2[0], c1 = b2[1], c2 = b2[2];

  const float L2E  = 1.44269504088896340736f;   // log2(e)
  const float nL2E = -L2E;
  const float vu  = v0 * u0 + v1 * u1 + v2 * u2;          // v . u
  const float vc  = v0 * c0 + v1 * c1 + v2 * c2;          // v . c
  const float vm0 = v0 * M00 + v1 * M10 + v2 * M20;       // (v M)_k
  const float vm1 = v0 * M01 + v1 * M11 + v2 * M21;
  const float vm2 = v0 * M02 + v1 * M12 + v2 * M22;
  const float bvc = bb1 + vc;
  const float cn  = nL2E * vu;                            // on-path fma coeff

  float p0 = 0.f, p1 = 0.f, p2 = 0.f;

  // Rolling 4-wide buffers: cb = a[tb..tb+3], nb = a[tb+4..tb+7]
  v4f cb = *(const v4f*)(a);
  v4f nb = *(const v4f*)(a + 4);
  float g = nL2E * (cb[0] + bb1);   // g_0 = -log2e * s_0  (p_0 = 0)

  auto STEP = [&](float an) {
    // ---- critical path ----
    float e = __builtin_amdgcn_exp2f(g);        // exp(-s_t)
    float h = __builtin_amdgcn_rcpf(1.0f + e);  // sigmoid (1 ulp rcp)
    // ---- off-path (depend only on previous p / loaded a) ----
    float q  = fmaf(vm0, p0, fmaf(vm1, p1, fmaf(vm2, p2, an + bvc)));
    float t0 = fmaf(M00, p0, fmaf(M01, p1, fmaf(M02, p2, c0)));
    float t1 = fmaf(M10, p0, fmaf(M11, p1, fmaf(M12, p2, c1)));
    float t2 = fmaf(M20, p0, fmaf(M21, p1, fmaf(M22, p2, c2)));
    // ---- one fma back onto the critical path ----
    g  = fmaf(cn, h, nL2E * q);
    p0 = fmaf(u0, h, t0);
    p1 = fmaf(u1, h, t1);
    p2 = fmaf(u2, h, t2);
  };

  for (int tb = 0; tb < T_LEN; tb += 4) {
    int pf = tb + 8;
    if (pf > T_LEN - 4) pf = T_LEN - 4;         // clamp: stay in bounds; the
    v4f fb = *(const v4f*)(a + pf);             // misfed value only reaches a
                                                // g that is never consumed
    STEP(cb[1]);    // needs a[tb+1]
    STEP(cb[2]);    // a[tb+2]
    STEP(cb[3]);    // a[tb+3]
    STEP(nb[0]);    // a[tb+4]
    cb = nb;
    nb = fb;
  }

  out[0] = p0;
  out[1] = p1;
  out[2] = p2;
}

// ---------------------------------------------------------------------------
extern "C" void kernel_launch(void* const* d_in, const int* in_sizes, int n_in,
                              void* d_out, int out_size, void* d_ws, size_t ws_size,
                              hipStream_t stream) {
  const float* doc = (const float*)d_in[0];  // (T, E)
  const float* w1  = (const float*)d_in[1];  // (1, E+O)
  const float* b1  = (const float*)d_in[2];  // (1,)
  const float* w2  = (const float*)d_in[3];  // (3, 4)
  const float* b2  = (const float*)d_in[4];  // (3,)
  float* out = (float*)d_out;                // 3 floats
  float* a   = (float*)d_ws;                 // T floats = 128 KB scratch

  // 2048 waves, 16 rows each: 256 blocks x 8 waves
  rnn_gemv_wmma<<<T_LEN / 128, 256, 0, stream>>>(doc, w1, a);
  // Serial tail depends on `a`; same stream serializes the two kernels.
  rnn_serial<<<1, 32, 0, stream>>>(a, w1, b1, w2, b2, out);
}